// MatchingLayer_14465449853028
// MI455X (gfx1250) — compile-verified
//
#include <hip/hip_runtime.h>

// Problem constants (match reference)
#define BB 8
#define PP 64
#define LL 128
#define HH 768
#define WW 16
#define F3H (3 * HH)          // 2304
#define NPAIR (BB * PP)       // 512
#define NEG_INF (-1e30f)

#define WPAD_ROWS 16          // N dim padded 4 -> 16 for WMMA
#define WPAD_ELEMS (WPAD_ROWS * F3H)  // 36864 floats

typedef __attribute__((ext_vector_type(2))) float v2f;
typedef __attribute__((ext_vector_type(4))) float v4f;
typedef __attribute__((ext_vector_type(8))) float v8f;

// ---------------------------------------------------------------------------
// Kernel 0: pad weight [4, 2304] -> wpad [16, 2304] (rows 4..15 zero).
// Removes all masking from the WMMA hot loop.
// ---------------------------------------------------------------------------
__global__ __launch_bounds__(256) void pad_weight_kernel(
    const float* __restrict__ weight,  // [4, 3H]
    float*       __restrict__ wpad)    // [16, 3H]
{
    const int i = blockIdx.x * 256 + threadIdx.x;   // 0 .. WPAD_ELEMS-1
    if (i < WPAD_ELEMS) {
        const int row = i / F3H;
        wpad[i] = (row < 4) ? weight[i] : 0.0f;
    }
}

// ---------------------------------------------------------------------------
// Kernel 1: encode. One block per (b,p) pair. 192 threads; thread t owns
// channels [4t, 4t+4). All loop bounds are uniform per block -> no divergence;
// loads are fully coalesced b128 (192 lanes x float4 = one 3KB channel row).
// feat row layout: [S(768) | E(768) | R(768)]  -> feat[NPAIR][2304]
// ---------------------------------------------------------------------------
__global__ __launch_bounds__(192) void encode_kernel(
    const float* __restrict__ table,   // [B, L, L, H]
    const int*   __restrict__ pairs,   // [B, P, 4]
    float*       __restrict__ feat)    // [NPAIR, 3H] (workspace)
{
    const int bp = blockIdx.x;            // 0..511
    const int b  = bp >> 6;               // bp / P
    const int t  = threadIdx.x;           // 0..191
    const int ch = t << 2;                // channel base (0..764)

    const int4 pr = ((const int4*)pairs)[bp];
    const int s0 = pr.x, e0 = pr.y, s1 = pr.z, e1 = pr.w;

    const float* tb = table + (size_t)b * LL * LL * HH;

    // S = table[b, s0+1, s1+1, :]
    const v4f Sv = *(const v4f*)(tb + ((size_t)(s0 + 1) * LL + (s1 + 1)) * HH + ch);
    // E = table[b, e0, e1, :]
    const v4f Ev = *(const v4f*)(tb + ((size_t)e0 * LL + e1) * HH + ch);

    // R = max over rows s0+1..e0, cols s1+1..e1 (>=1 cell each guaranteed)
    v4f Rv;
    Rv.x = NEG_INF; Rv.y = NEG_INF; Rv.z = NEG_INF; Rv.w = NEG_INF;
    const int nr = e0 - s0;
    const int nc = e1 - s1;
    for (int r = 0; r < nr; ++r) {
        const float* rowp = tb + ((size_t)(s0 + 1 + r) * LL + (s1 + 1)) * HH + ch;
        for (int c = 0; c < nc; ++c) {
            const v4f v = *(const v4f*)(rowp + (size_t)c * HH);
            Rv.x = fmaxf(Rv.x, v.x);
            Rv.y = fmaxf(Rv.y, v.y);
            Rv.z = fmaxf(Rv.z, v.z);
            Rv.w = fmaxf(Rv.w, v.w);
        }
    }

    float* frow = feat + (size_t)bp * F3H;
    *(v4f*)(frow + ch)            = Sv;
    *(v4f*)(frow + HH + ch)       = Ev;
    *(v4f*)(frow + 2 * HH + ch)   = Rv;
}

// ---------------------------------------------------------------------------
// Kernel 2: projection  out[512,4] = feat[512,2304] x wpad^T[2304,16] + bias
// via V_WMMA_F32_16X16X4_F32 (f32 in, f32 acc -> exact). One wave per tile of
// 16 pairs. Zero-padded B means no masking in the loop; 4 independent
// accumulators (K interleaved by 16) break the WMMA->WMMA RAW chain.
//
// Layouts (ISA 7.12.2, 32-bit, wave32):
//   A 16x4 : VGPR j, lane l -> row (l&15), K = (l>>4)*2 + j      -> float2/lane
//   B 4x16 : VGPR j, lane l -> col (l&15), K = (l>>4)*2 + j      -> float2/lane
//   D 16x16: VGPR v, lane l -> M = v + 8*(l>>4), N = (l&15)
// ---------------------------------------------------------------------------
__global__ __launch_bounds__(32) void proj_wmma_kernel(
    const float* __restrict__ feat,    // [NPAIR, 3H]
    const float* __restrict__ wpad,    // [16, 3H] zero-padded
    const float* __restrict__ bias,    // [4]
    float*       __restrict__ out)     // [NPAIR, 4]
{
    const int tile = blockIdx.x;          // 0..31 -> pairs [tile*16, tile*16+16)
    const int lane = threadIdx.x;         // 0..31
    const int half = lane >> 4;           // 0 or 1
    const int l15  = lane & 15;

    const float* arow = feat + (size_t)(tile * 16 + l15) * F3H + half * 2;
    const float* brow = wpad + (size_t)l15 * F3H + half * 2;

    v8f a0 = {}, a1 = {}, a2 = {}, a3 = {};
    for (int k0 = 0; k0 < F3H; k0 += 16) {
        const v2f av0 = *(const v2f*)(arow + k0);
        const v2f bv0 = *(const v2f*)(brow + k0);
        const v2f av1 = *(const v2f*)(arow + k0 + 4);
        const v2f bv1 = *(const v2f*)(brow + k0 + 4);
        const v2f av2 = *(const v2f*)(arow + k0 + 8);
        const v2f bv2 = *(const v2f*)(brow + k0 + 8);
        const v2f av3 = *(const v2f*)(arow + k0 + 12);
        const v2f bv3 = *(const v2f*)(brow + k0 + 12);
        // (neg_a, A, neg_b, B, c_mod, C, reuse_a, reuse_b)
        a0 = __builtin_amdgcn_wmma_f32_16x16x4_f32(false, av0, false, bv0, (short)0, a0, false, false);
        a1 = __builtin_amdgcn_wmma_f32_16x16x4_f32(false, av1, false, bv1, (short)0, a1, false, false);
        a2 = __builtin_amdgcn_wmma_f32_16x16x4_f32(false, av2, false, bv2, (short)0, a2, false, false);
        a3 = __builtin_amdgcn_wmma_f32_16x16x4_f32(false, av3, false, bv3, (short)0, a3, false, false);
    }

    // Combine partials; lanes with N<4 own output column N; VGPR v -> pair M.
    if (l15 < 4) {
        const float bi = bias[l15];
        #pragma unroll
        for (int v = 0; v < 8; ++v) {
            const int m = v + half * 8;
            out[(size_t)(tile * 16 + m) * 4 + l15] =
                ((a0[v] + a1[v]) + (a2[v] + a3[v])) + bi;
        }
    }
}

extern "C" void kernel_launch(void* const* d_in, const int* in_sizes, int n_in,
                              void* d_out, int out_size, void* d_ws, size_t ws_size,
                              hipStream_t stream) {
    (void)in_sizes; (void)n_in; (void)out_size; (void)ws_size;
    const float* table  = (const float*)d_in[0];
    const int*   pairs  = (const int*)d_in[1];
    const float* weight = (const float*)d_in[2];
    const float* bias   = (const float*)d_in[3];
    float*       out    = (float*)d_out;

    // Workspace layout: feat [512*2304 f32] | wpad [16*2304 f32]  (~4.87 MB)
    float* feat = (float*)d_ws;
    float* wpad = feat + (size_t)NPAIR * F3H;

    pad_weight_kernel<<<(WPAD_ELEMS + 255) / 256, 256, 0, stream>>>(weight, wpad);
    encode_kernel<<<NPAIR, 192, 0, stream>>>(table, pairs, feat);
    proj_wmma_kernel<<<NPAIR / 16, 32, 0, stream>>>(feat, wpad, bias, out);
}